// AlignSnippets_48198122996101
// MI455X (gfx1250) — compile-verified
//
#include <hip/hip_runtime.h>

#define TT 64
#define CC 16
#define ROW_FLOATS (TT * CC)        // 1024 floats = 4096 B per snippet row
#define ROW_VEC4   (ROW_FLOATS / 4) // 256 x float4 chunks -> 256 threads/block
#define ROWS_PER_BLOCK 16           // batch rows to amortize block dispatch

typedef float f32x4 __attribute__((ext_vector_type(4)));
typedef int   i32x4 __attribute__((__vector_size__(4 * sizeof(int))));

#if defined(__gfx1250__) && __has_builtin(__builtin_amdgcn_global_load_async_to_lds_b128)
#define USE_ASYNC_LDS 1
#else
#define USE_ASYNC_LDS 0
#endif

__device__ __forceinline__ void async_row_load(const f32x4* gsrc, f32x4* lbuf, int tid)
{
#if USE_ASYNC_LDS
    __builtin_amdgcn_global_load_async_to_lds_b128(
        (__attribute__((address_space(1))) i32x4*)(gsrc + tid),
        (__attribute__((address_space(3))) i32x4*)(lbuf + tid),
        /*offset=*/0, /*cpol=*/0);
#else
    lbuf[tid] = gsrc[tid];
#endif
}

__device__ __forceinline__ void wait_async_le(int n)
{
#if USE_ASYNC_LDS
#if __has_builtin(__builtin_amdgcn_s_wait_asynccnt)
    if (n == 0) __builtin_amdgcn_s_wait_asynccnt(0);
    else        __builtin_amdgcn_s_wait_asynccnt(1);
#else
    if (n == 0) asm volatile("s_wait_asynccnt 0x0" ::: "memory");
    else        asm volatile("s_wait_asynccnt 0x1" ::: "memory");
#endif
#endif
}

// Each block streams ROWS_PER_BLOCK snippet rows through a double-buffered
// LDS staging area: the async DMA for row r+1 is in flight while row r is
// rolled (LDS read with shifted index) and stored non-temporally.
__global__ __launch_bounds__(ROW_VEC4) void AlignSnippets_roll_kernel(
    const float* __restrict__ snippets,   // [N, T, C]
    const int*   __restrict__ times,      // [N]
    const int*   __restrict__ labels,     // [N], 1..K valid, 0 = unassigned
    const int*   __restrict__ offsets,    // [K]
    float* __restrict__ out_aligned,      // [N, T, C]
    float* __restrict__ out_times,        // [N]
    int N, int K)
{
    __shared__ f32x4 buf[2][ROW_VEC4];    // 2 x 4 KB staging buffers

    const int tid  = threadIdx.x;
    const long long row0 = (long long)blockIdx.x * ROWS_PER_BLOCK;
    int nrows = N - (int)row0;
    if (nrows > ROWS_PER_BLOCK) nrows = ROWS_PER_BLOCK;
    if (nrows <= 0) return;

    const int t_out = tid >> 2;           // 0..63
    const int c4    = tid & 3;            // 0..3

    // Prime the pipeline with row 0 of this block.
    async_row_load((const f32x4*)(snippets + (size_t)row0 * ROW_FLOATS), buf[0], tid);

    for (int r = 0; r < nrows; ++r) {
        const long long n = row0 + r;

        // Launch DMA for the next row into the other buffer (overlaps stores).
        const bool have_next = (r + 1) < nrows;
        if (have_next) {
            async_row_load((const f32x4*)(snippets + (size_t)(n + 1) * ROW_FLOATS),
                           buf[(r + 1) & 1], tid);
            wait_async_le(1);             // oldest (row r) load retired; row r+1 in flight
        } else {
            wait_async_le(0);
        }
        __syncthreads();                  // all waves' portions of row r are in LDS

        const int  lab   = labels[n];
        const bool valid = (lab >= 1) && (lab <= K);
        const int  shift = valid ? offsets[lab - 1] : 0;

        // torch.roll: out[t] = in[(t - shift) mod T]; mask is exact for any sign.
        const int t_in = (t_out - shift) & (TT - 1);

        f32x4 v = {0.f, 0.f, 0.f, 0.f};
        if (valid) v = buf[r & 1][(t_in << 2) | c4];   // ds_load_b128

        // Output is written once and never re-read: stream it past the caches.
        __builtin_nontemporal_store(
            v, (f32x4*)(out_aligned + (size_t)n * ROW_FLOATS) + tid);

        if (tid == 0) {
            out_times[n] = valid ? (float)(times[n] - shift) : 0.0f;
        }
        __syncthreads();                  // buffer (r&1) free before r+2 prefetch reuses it
    }
}

extern "C" void kernel_launch(void* const* d_in, const int* in_sizes, int n_in,
                              void* d_out, int out_size, void* d_ws, size_t ws_size,
                              hipStream_t stream)
{
    const float* snippets = (const float*)d_in[0];
    const int*   times    = (const int*)d_in[1];
    const int*   labels   = (const int*)d_in[2];
    const int*   offsets  = (const int*)d_in[3];

    const int N = in_sizes[1];            // length of times == N
    const int K = in_sizes[3];            // length of offsets == K

    float* out_aligned = (float*)d_out;                          // N*T*C floats
    float* out_times   = (float*)d_out + (size_t)N * ROW_FLOATS; // then N floats

    const unsigned blocks = (unsigned)((N + ROWS_PER_BLOCK - 1) / ROWS_PER_BLOCK);
    AlignSnippets_roll_kernel<<<dim3(blocks), dim3(ROW_VEC4), 0, stream>>>(
        snippets, times, labels, offsets, out_aligned, out_times, N, K);
}